// GCNConv_28355374088416
// MI455X (gfx1250) — compile-verified
//
#include <hip/hip_runtime.h>

typedef float v2f __attribute__((ext_vector_type(2)));
typedef float v8f __attribute__((ext_vector_type(8)));

#define NN 8192
#define D 128

// ---------------- Kernel A: deg row-sums -> dinv = rsqrt(deg) (0 if deg<=0) ----
// adj is pure streaming data (268 MB > 192 MB L2): load non-temporal.
__global__ __launch_bounds__(256) void k_deg(const float* __restrict__ adj,
                                             float* __restrict__ dinv) {
  __shared__ float red[256];
  const int row = blockIdx.x;
  const float* r = adj + (size_t)row * NN;
  float s = 0.f;
  for (int j = threadIdx.x; j < NN; j += 256)
    s += __builtin_nontemporal_load(r + j);
  red[threadIdx.x] = s;
  __syncthreads();
  for (int off = 128; off > 0; off >>= 1) {
    if (threadIdx.x < off) red[threadIdx.x] += red[threadIdx.x + off];
    __syncthreads();
  }
  if (threadIdx.x == 0) {
    float d = red[0];
    dinv[row] = d > 0.f ? rsqrtf(d) : 0.f;
  }
}

// ---------------- Kernel B: hs = dinv[row] * (x @ W), stored in WMMA-B fragment
// layout. Fragment (ks = k/4, tile = n/16): 32 lanes x {vgpr0,vgpr1} float2,
// element (K = r + 2*(lane>=16), N = lane&15)  ->  float2 index
// ((ks*8 + tile)*32 + lane), component r.
__global__ __launch_bounds__(128) void k_proj(const float* __restrict__ x,
                                              const float* __restrict__ w,
                                              const float* __restrict__ dinv,
                                              float* __restrict__ bfrag) {
  __shared__ float xs[D];
  const int row = blockIdx.x;
  const int col = threadIdx.x;
  xs[col] = x[(size_t)row * D + col];
  __syncthreads();
  float acc = 0.f;
#pragma unroll 8
  for (int k = 0; k < D; ++k) acc += xs[k] * w[k * D + col];
  acc *= dinv[row];

  const int ks = row >> 2, within = row & 3;
  const int r = within & 1, half = within >> 1;
  const int lane = half * 16 + (col & 15);
  const int tile = col >> 4;
  bfrag[(((size_t)ks * 8 + tile) * 32 + lane) * 2 + r] = acc;
}

// ---------------- Kernel C: out = dinv[i] * (adj @ hs) + bias, via fp32 WMMA ---
// One wave per 16-row x 128-col output stripe; 4 waves / block.
// adj streamed non-temporally; bfrag (4 MB) stays hot in L2.
__global__ __launch_bounds__(128) void k_agg(const float* __restrict__ adj,
                                             const float* __restrict__ bfrag,
                                             const float* __restrict__ dinv,
                                             const float* __restrict__ bias,
                                             float* __restrict__ out) {
  const int lane = threadIdx.x & 31;
  const int wave = threadIdx.x >> 5;
  const int row0 = blockIdx.x * 64 + wave * 16;
  const int lo = lane & 15, hi = lane >> 4;

  // A fragment source: lane holds adj[row0+lo][4*ks + 2*hi + {0,1}]
  const v2f* arow = (const v2f*)(adj + (size_t)(row0 + lo) * NN) + hi;
  // B fragments: one coalesced float2 per lane per (ks, tile)
  const v2f* bf = (const v2f*)bfrag + lane;

  v8f c[8] = {};

  for (int ks = 0; ks < NN / 4; ++ks) {
    const v2f a = __builtin_nontemporal_load(arow + (size_t)ks * 2);
    const v2f* bks = bf + (size_t)ks * (8 * 32);
#pragma unroll
    for (int t = 0; t < 8; ++t) {
      const v2f b = bks[t * 32];
      c[t] = __builtin_amdgcn_wmma_f32_16x16x4_f32(
          /*neg_a=*/false, a, /*neg_b=*/false, b,
          /*c_mod=*/(short)0, c[t], /*reuse_a=*/false, /*reuse_b=*/false);
    }
  }

  // Epilogue: C layout -> VGPR v holds M = v + 8*hi, N = lo (+ 16*tile)
#pragma unroll
  for (int t = 0; t < 8; ++t) {
    const int col = t * 16 + lo;
    const float bv = bias[col];
#pragma unroll
    for (int v = 0; v < 8; ++v) {
      const int row = row0 + hi * 8 + v;
      out[(size_t)row * D + col] = dinv[row] * c[t][v] + bv;
    }
  }
}

extern "C" void kernel_launch(void* const* d_in, const int* in_sizes, int n_in,
                              void* d_out, int out_size, void* d_ws, size_t ws_size,
                              hipStream_t stream) {
  const float* x    = (const float*)d_in[0];   // [8192,128]
  const float* adj  = (const float*)d_in[1];   // [8192,8192]
  const float* w    = (const float*)d_in[2];   // [128,128]
  const float* bias = (const float*)d_in[3];   // [128]
  float* out = (float*)d_out;                  // [8192,128]

  float* dinv  = (float*)d_ws;                 // 8192 floats
  float* bfrag = dinv + NN;                    // 8192*128 floats (4 MB), fragment layout

  k_deg <<<NN, 256, 0, stream>>>(adj, dinv);
  k_proj<<<NN, 128, 0, stream>>>(x, w, dinv, bfrag);
  k_agg <<<NN / 64, 128, 0, stream>>>(adj, bfrag, dinv, bias, out);
}